// ClassECELoss_47923245089173
// MI455X (gfx1250) — compile-verified
//
#include <hip/hip_runtime.h>
#include <stdint.h>

#define NBINS 15
#define MAXC 1024
#define WAVES_PER_WG 8
#define BLOCK (WAVES_PER_WG * 32)
#define GRID_ACC 256

typedef float v2f __attribute__((ext_vector_type(2)));
typedef float v8f __attribute__((ext_vector_type(8)));

// ---------------------------------------------------------------------------
// Async-stage one row (C <= 1024 fp32) from global into LDS using the CDNA5
// async-DMA path (ASYNCcnt-tracked, no VGPR data movement).
// ---------------------------------------------------------------------------
__device__ __forceinline__ void async_stage_row(const float* __restrict__ gsrc,
                                                uint32_t lds_base, int lane,
                                                int nchunks /* 16B chunks */) {
#pragma unroll
  for (int k = 0; k < 8; ++k) {
    int chunk = lane + (k << 5);
    if (chunk < nchunks) {
      uint32_t la = lds_base + (uint32_t)(chunk << 4);
      uint64_t ga = (uint64_t)(uintptr_t)gsrc + (uint32_t)(chunk << 4);
      asm volatile("global_load_async_to_lds_b128 %0, %1, off"
                   :: "v"(la), "v"(ga)
                   : "memory");
    }
  }
}

__device__ __forceinline__ void wait_async0() {
  asm volatile("s_wait_asynccnt 0" ::: "memory");
}

// bin + scatter one probability (replicates searchsorted(linspace(0,1,16),p,'left')-1)
__device__ __forceinline__ void bin_scatter(float p, int j, int lab,
                                            float* __restrict__ s_diff) {
  int b = (int)(p * 15.0f);
  if (b > 14) b = 14;
  if (p <= (float)b * (1.0f / 15.0f)) {
    b -= 1;                                   // p == boundary -> lower bin
  } else if (p > (float)(b + 1) * (1.0f / 15.0f)) {
    b += 1; if (b > 14) b = 14;
  }
  if (b >= 0) {                               // p == 0 -> dropped (bin -1)
    float contrib = p - ((j == lab) ? 1.0f : 0.0f);
    atomicAdd(&s_diff[j * NBINS + b], contrib);    // ds_add_f32
  }
}

// ---------------------------------------------------------------------------
// Accumulation kernel: one wave per row; (class,bin) diff accumulator in LDS.
// ---------------------------------------------------------------------------
__global__ __launch_bounds__(BLOCK) void ece_accum(
    const float* __restrict__ logits, const int* __restrict__ labels,
    int N, int C,
    float* __restrict__ g_diff, float* __restrict__ g_corr,
    float* __restrict__ g_tot) {

  __shared__ float s_diff[MAXC * NBINS];                         // 61440 B
  __shared__ __align__(16) float s_stage[WAVES_PER_WG][2][1024]; // 65536 B

  const int wv = threadIdx.x >> 5;
  const int lane = threadIdx.x & 31;
  const int nbe = C * NBINS;

  for (int i = threadIdx.x; i < nbe; i += BLOCK) s_diff[i] = 0.0f;
  __syncthreads();

  const int nchunks = (C + 3) >> 2;  // 16B chunks to stage
  const int nch4 = C >> 2;           // full float4 chunks
  const int ctail = C & ~3;
  const uint32_t base0 = (uint32_t)(uintptr_t)&s_stage[wv][0][0];
  const uint32_t base1 = (uint32_t)(uintptr_t)&s_stage[wv][1][0];

  int cur = blockIdx.x * WAVES_PER_WG + wv;
  const int wstride = gridDim.x * WAVES_PER_WG;

  if (cur < N) async_stage_row(logits + (size_t)cur * C, base0, lane, nchunks);
  int buf = 0;

  while (cur < N) {
    const int nxt = cur + wstride;
    wait_async0();  // current buffer fully landed in LDS
    if (nxt < N)    // prefetch next row into the other buffer
      async_stage_row(logits + (size_t)nxt * C, buf ? base0 : base1, lane, nchunks);

    float* row = &s_stage[wv][buf][0];
    float4* row4 = (float4*)row;     // 16B-aligned -> ds_*_b128

    // ---- pass A: row max + first-argmax (matches jnp.argmax tie-break) ----
    float bm = -3.402823466e38f;
    int bi = 0;
#pragma unroll
    for (int k = 0; k < 8; ++k) {
      int ch = lane + (k << 5);
      if (ch < nch4) {
        float4 v = row4[ch];
        int j = ch << 2;
        if (v.x > bm) { bm = v.x; bi = j; }
        if (v.y > bm) { bm = v.y; bi = j + 1; }
        if (v.z > bm) { bm = v.z; bi = j + 2; }
        if (v.w > bm) { bm = v.w; bi = j + 3; }
      }
    }
    for (int j = ctail + lane; j < C; j += 32) {
      float v = row[j];
      if (v > bm) { bm = v; bi = j; }
    }
    for (int off = 16; off > 0; off >>= 1) {
      float om = __shfl_xor(bm, off, 32);
      int   oi = __shfl_xor(bi, off, 32);
      if (om > bm || (om == bm && oi < bi)) { bm = om; bi = oi; }
    }

    // ---- pass B: exp(x - max) written back over staged row; sum-reduce ----
    float se = 0.0f;
#pragma unroll
    for (int k = 0; k < 8; ++k) {
      int ch = lane + (k << 5);
      if (ch < nch4) {
        float4 v = row4[ch];
        v.x = __expf(v.x - bm);
        v.y = __expf(v.y - bm);
        v.z = __expf(v.z - bm);
        v.w = __expf(v.w - bm);
        row4[ch] = v;
        se += (v.x + v.y) + (v.z + v.w);
      }
    }
    for (int j = ctail + lane; j < C; j += 32) {
      float e = __expf(row[j] - bm);
      row[j] = e;
      se += e;
    }
    for (int off = 16; off > 0; off >>= 1) se += __shfl_xor(se, off, 32);
    const float inv = 1.0f / se;

    const int lab = labels[cur];

    // ---- pass C: bin + scatter (conf - label_match) into LDS accumulator ----
#pragma unroll
    for (int k = 0; k < 8; ++k) {
      int ch = lane + (k << 5);
      if (ch < nch4) {
        float4 v = row4[ch];
        int j = ch << 2;
        bin_scatter(v.x * inv, j + 0, lab, s_diff);
        bin_scatter(v.y * inv, j + 1, lab, s_diff);
        bin_scatter(v.z * inv, j + 2, lab, s_diff);
        bin_scatter(v.w * inv, j + 3, lab, s_diff);
      }
    }
    for (int j = ctail + lane; j < C; j += 32)
      bin_scatter(row[j] * inv, j, lab, s_diff);

    if (lane == 0) {
      atomicAdd(&g_corr[lab], (bi == lab) ? 1.0f : 0.0f);
      atomicAdd(&g_tot[lab], 1.0f);
    }

    cur = nxt;
    buf ^= 1;
  }

  __syncthreads();
  // flush per-WG accumulator; skip exactly-zero cells (most high bins empty)
  for (int i = threadIdx.x; i < nbe; i += BLOCK) {
    float v = s_diff[i];
    if (v != 0.0f) atomicAdd(&g_diff[i], v);
  }
}

// ---------------------------------------------------------------------------
__global__ void zero_ws(float* __restrict__ p, int n) {
  int i = blockIdx.x * blockDim.x + threadIdx.x;
  if (i < n) p[i] = 0.0f;
}

// Branch-free guarded |diff| load (keeps EXEC all-ones around WMMA).
__device__ __forceinline__ float ldabs(const float* __restrict__ g, int base,
                                       int b, bool cok) {
  int bb = (b < NBINS) ? b : 0;
  float v = fabsf(g[base + bb]);
  return (cok && b < NBINS) ? v : 0.0f;
}

// ---------------------------------------------------------------------------
// Finalize: per_class_sce[c] = (sum_b |diff[c][b]|)/N  via chained
// V_WMMA_F32_16X16X4_F32 (A = 16x16 tile of |diff| fed as 4 K=4 slabs,
// B = ones), plus classes_acc = corr/tot.
// ---------------------------------------------------------------------------
__global__ __launch_bounds__(256) void ece_finalize(
    const float* __restrict__ g_diff, const float* __restrict__ g_corr,
    const float* __restrict__ g_tot, float* __restrict__ out, int C, int N) {

  const int wv = threadIdx.x >> 5;
  const int lane = threadIdx.x & 31;

  // accuracy part (divergence reconverges before WMMA)
  int c = blockIdx.x * blockDim.x + threadIdx.x;
  if (c < C) out[C + c] = g_corr[c] / g_tot[c];

  const int tile = blockIdx.x * (256 / 32) + wv;
  const int ntiles = (C + 15) >> 4;
  if (tile >= ntiles) return;  // wave-uniform

  // A-matrix 16x4 f32 layout: lanes 0-15 -> M=lane, K={0,1}; lanes 16-31 ->
  // M=lane-16, K={2,3}.  (ISA 7.12.2)
  const int mrow = lane & 15;
  const int cls = (tile << 4) + mrow;
  const bool cok = cls < C;
  const int cbase = (cok ? cls : 0) * NBINS;
  const int khalf = (lane >> 4) << 1;  // 0 or 2

  const v2f Bones = {1.0f, 1.0f};      // all-ones: layout-invariant
  v8f acc = {};
#pragma unroll
  for (int q = 0; q < 4; ++q) {        // bins q*4 .. q*4+3 (bin 15 padded 0)
    int b0 = (q << 2) + khalf;
    v2f A;
    A.x = ldabs(g_diff, cbase, b0, cok);
    A.y = ldabs(g_diff, cbase, b0 + 1, cok);
    acc = __builtin_amdgcn_wmma_f32_16x16x4_f32(
        false, A, false, Bones, (short)0, acc, false, false);
  }

  // D layout: VGPR r holds M=r (lanes 0-15) / M=r+8 (lanes 16-31), any N.
  if (lane == 0 || lane == 16) {
    int mbase = (tile << 4) + ((lane >> 4) << 3);
#pragma unroll
    for (int r = 0; r < 8; ++r) {
      int cc = mbase + r;
      if (cc < C) out[cc] = acc[r] / (float)N;   // per_class_sce
    }
  }
}

// ---------------------------------------------------------------------------
extern "C" void kernel_launch(void* const* d_in, const int* in_sizes, int n_in,
                              void* d_out, int out_size, void* d_ws, size_t ws_size,
                              hipStream_t stream) {
  const float* logits = (const float*)d_in[0];
  const int*   labels = (const int*)d_in[1];
  const int N = in_sizes[1];
  const int C = in_sizes[0] / in_sizes[1];

  float* ws     = (float*)d_ws;
  float* g_diff = ws;                 // C*NBINS
  float* g_corr = ws + C * NBINS;     // C
  float* g_tot  = g_corr + C;         // C
  const int nz = C * NBINS + 2 * C;

  zero_ws<<<(nz + 255) / 256, 256, 0, stream>>>(ws, nz);
  ece_accum<<<GRID_ACC, BLOCK, 0, stream>>>(logits, labels, N, C,
                                            g_diff, g_corr, g_tot);

  const int ntiles = (C + 15) >> 4;
  int fblocks = (ntiles + 7) / 8;
  int ablocks = (C + 255) / 256;
  int blocks = fblocks > ablocks ? fblocks : ablocks;
  ece_finalize<<<blocks, 256, 0, stream>>>(g_diff, g_corr, g_tot,
                                           (float*)d_out, C, N);
}